// JNGOOptimizer_70377334112777
// MI455X (gfx1250) — compile-verified
//
#include <hip/hip_runtime.h>
#include <hip/hip_bf16.h>
#include <math.h>

// ---------------------------------------------------------------------------
// LM pose-hypothesis refinement for MI455X (gfx1250, wave32).
//   B=8 batches, M=144 hypotheses (first 16 scored), N=500 anchor points.
//   Per-wave: V_WMMA_F32_16X16X4_F32 computes (16 hyp rows x K=4) @ (4 x 16 pts)
//   for the x/y/z camera rows in 3 WMMAs; each lane then owns 8 (hyp,point)
//   pairs and does the bilinear feature gather + channel SSD in FP32.
//   r-loop fully unrolled so WMMA accumulator reads are direct VGPR accesses
//   (no M0/v_movrels indexed addressing).
// ---------------------------------------------------------------------------

#define B_SZ   8
#define M_HYP  144
#define N_PTS  500

typedef float v2f __attribute__((ext_vector_type(2)));
typedef float v8f __attribute__((ext_vector_type(8)));

// ---------------- deterministic PRNG (stand-in for jax threefry) -----------
__device__ __forceinline__ unsigned pcg_hash(unsigned x) {
  x = x * 747796405u + 2891336453u;
  unsigned w = ((x >> ((x >> 28u) + 4u)) ^ x) * 277803737u;
  return (w >> 22u) ^ w;
}
__device__ __forceinline__ float hash_normal(unsigned seed) {
  unsigned a = pcg_hash(seed * 2u + 0x9E3779B9u);
  unsigned b = pcg_hash(seed * 2u + 0x85EBCA6Bu);
  float u1 = ((float)a + 1.0f) * 2.3283064365386963e-10f;  // (0,1]
  float u2 = (float)b * 2.3283064365386963e-10f;           // [0,1)
  return sqrtf(-2.0f * logf(u1)) * cosf(6.283185307179586f * u2);
}

// ---------------- small matrix helpers -------------------------------------
__device__ __forceinline__ void mat3mul(float* o, const float* a, const float* b) {
  for (int i = 0; i < 3; ++i)
    for (int j = 0; j < 3; ++j)
      o[i * 3 + j] = a[i * 3 + 0] * b[0 * 3 + j] +
                     a[i * 3 + 1] * b[1 * 3 + j] +
                     a[i * 3 + 2] * b[2 * 3 + j];
}
__device__ __forceinline__ void mat4mul(float* o, const float* a, const float* b) {
  for (int i = 0; i < 4; ++i)
    for (int j = 0; j < 4; ++j) {
      float s = 0.0f;
      for (int k = 0; k < 4; ++k) s += a[i * 4 + k] * b[k * 4 + j];
      o[i * 4 + j] = s;
    }
}

__device__ void se3_exp(const float* xi, float* T) {
  const float tx = xi[0], ty = xi[1], tz = xi[2];
  const float wx = xi[3], wy = xi[4], wz = xi[5];
  float theta = fmaxf(sqrtf(wx * wx + wy * wy + wz * wz), 1e-8f);
  float kx = wx / theta, ky = wy / theta, kz = wz / theta;
  float K[9] = {0.0f, -kz, ky, kz, 0.0f, -kx, -ky, kx, 0.0f};
  float KK[9];
  mat3mul(KK, K, K);
  float st = sinf(theta), ct = 1.0f - cosf(theta);
  float a = ct / theta, bb = 1.0f - st / theta;
  float R[9], V[9];
  for (int i = 0; i < 9; ++i) {
    float e = (i == 0 || i == 4 || i == 8) ? 1.0f : 0.0f;
    R[i] = e + st * K[i] + ct * KK[i];
    V[i] = e + a * K[i] + bb * KK[i];
  }
  float t0 = V[0] * tx + V[1] * ty + V[2] * tz;
  float t1 = V[3] * tx + V[4] * ty + V[5] * tz;
  float t2 = V[6] * tx + V[7] * ty + V[8] * tz;
  T[0] = R[0]; T[1] = R[1]; T[2]  = R[2]; T[3]  = t0;
  T[4] = R[3]; T[5] = R[4]; T[6]  = R[5]; T[7]  = t1;
  T[8] = R[6]; T[9] = R[7]; T[10] = R[8]; T[11] = t2;
  T[12] = 0.0f; T[13] = 0.0f; T[14] = 0.0f; T[15] = 1.0f;
}

__device__ void se3_log(const float* T, float* xi) {
  float cos_a = (T[0] + T[5] + T[10] - 1.0f) * 0.5f;
  cos_a = fminf(fmaxf(cos_a, -1.0f + 1e-7f), 1.0f - 1e-7f);
  float theta = acosf(cos_a);
  float th = fmaxf(theta, 1e-8f);
  float s = th / (2.0f * fmaxf(sinf(th), 1e-8f));
  float wx = (T[9] - T[6]) * s;
  float wy = (T[2] - T[8]) * s;
  float wz = (T[4] - T[1]) * s;
  if (fabsf(theta) < 1e-6f) { wx = 0.0f; wy = 0.0f; wz = 0.0f; }
  xi[0] = T[3]; xi[1] = T[7]; xi[2] = T[11];
  xi[3] = wx;   xi[4] = wy;   xi[5] = wz;
}

// ---------------- kernel: generate 144 hypotheses per batch ----------------
__global__ void gen_hyp(const float* __restrict__ Tpred, float* __restrict__ Tcur) {
  int idx = blockIdx.x * blockDim.x + threadIdx.x;
  if (idx >= B_SZ * M_HYP) return;
  int b = idx / M_HYP, m = idx % M_HYP;
  const float d2r = 0.017453292519943295f;
  float pitch = (-11.0f + 2.0f * (float)(m / 12)) * d2r;
  float yaw   = (-11.0f + 2.0f * (float)(m % 12)) * d2r;
  float xi[6];
  xi[0] = hash_normal(1000u + (unsigned)m * 3u + 0u);
  xi[1] = hash_normal(1000u + (unsigned)m * 3u + 1u);
  xi[2] = hash_normal(1000u + (unsigned)m * 3u + 2u);
  xi[3] = 0.0f; xi[4] = pitch; xi[5] = yaw;
  float dT[16];
  se3_exp(xi, dT);
  float Tp[16];
  for (int i = 0; i < 16; ++i) Tp[i] = Tpred[b * 16 + i];
  float To[16];
  mat4mul(To, dT, Tp);
  float* dst = Tcur + (size_t)idx * 16;
  for (int i = 0; i < 16; ++i) dst[i] = To[i];
}

__global__ void zero_res(float* __restrict__ res) {
  int i = blockIdx.x * blockDim.x + threadIdx.x;
  if (i < B_SZ * M_HYP) res[i] = 0.0f;
}

// ---------------- kernel: residuals (WMMA projection + gather SSD) ---------
// One wave per (batch, 16-point tile). A = 16 hypothesis rows (K=4), striped
// per the gfx1250 32-bit A layout (lanes 0-15 hold K=0,1; lanes 16-31 K=2,3);
// B = homogeneous points, same K striping. D: lane l, vgpr r -> hyp (8*half+r),
// point l.
__global__ __launch_bounds__(32)
void residual_kernel(const float* __restrict__ Tcur, const float* __restrict__ qf,
                     const float* __restrict__ rf, const float* __restrict__ qu,
                     const float* __restrict__ geo, const float* __restrict__ Kin,
                     float* __restrict__ res, int C, int H, int W, float scale) {
  __shared__ float smem[16][17];
  const int b = blockIdx.y;
  const int tile = blockIdx.x;
  const int lane = threadIdx.x;
  const int l = lane & 15;
  const int half = lane >> 4;
  const int k0 = half * 2;

  // B-matrix column: this lane's anchor point (clamped when padding past N).
  int n = tile * 16 + l;
  const bool valid = (n < N_PTS);
  const int ng = valid ? n : (N_PTS - 1);
  const float* gp = geo + ((size_t)b * N_PTS + ng) * 3;
  float ph[4] = {gp[0], gp[1], gp[2], 1.0f};
  v2f bm;
  bm.x = ph[k0]; bm.y = ph[k0 + 1];

  // A-matrix rows: hypothesis l, camera rows x/y/z, K-slice per lane half.
  const float* Tb = Tcur + ((size_t)b * M_HYP + l) * 16;
  v2f ax, ay, az;
  ax.x = Tb[0 * 4 + k0]; ax.y = Tb[0 * 4 + k0 + 1];
  ay.x = Tb[1 * 4 + k0]; ay.y = Tb[1 * 4 + k0 + 1];
  az.x = Tb[2 * 4 + k0]; az.y = Tb[2 * 4 + k0 + 1];

  v8f acc0 = {};
  v8f dx = __builtin_amdgcn_wmma_f32_16x16x4_f32(false, ax, false, bm, (short)0, acc0, false, false);
  v8f dy = __builtin_amdgcn_wmma_f32_16x16x4_f32(false, ay, false, bm, (short)0, acc0, false, false);
  v8f dz = __builtin_amdgcn_wmma_f32_16x16x4_f32(false, az, false, bm, (short)0, acc0, false, false);

  const float fx = Kin[b * 9 + 0] * scale;
  const float fy = Kin[b * 9 + 4] * scale;
  const float cx = Kin[b * 9 + 2] * scale;
  const float cy = Kin[b * 9 + 5] * scale;
  const float invN = 1.0f / (float)N_PTS;

  const size_t fs = (size_t)H * W;
  const float* qb = qf + (size_t)b * C * fs;
  const float* rb = rf + (size_t)b * C * fs;
  const float* ub = qu + (size_t)b * fs;

#pragma unroll
  for (int r = 0; r < 8; ++r) {
    const int m = half * 8 + r;
    float x = dx[r], y = dy[r], z = dz[r];   // constant subscripts after unroll
    z = fmaxf(z, 1e-6f);
    float u = fx * (x / z) + cx;
    float v = fy * (y / z) + cy;
    float gx = 2.0f * u / (float)(W - 1) - 1.0f;
    float gy = 2.0f * v / (float)(H - 1) - 1.0f;
    float xs = fminf(fmaxf(((gx + 1.0f) * (float)W - 1.0f) * 0.5f, 0.0f), (float)(W - 1));
    float ys = fminf(fmaxf(((gy + 1.0f) * (float)H - 1.0f) * 0.5f, 0.0f), (float)(H - 1));
    float x0 = floorf(xs), y0 = floorf(ys);
    float x1 = fminf(x0 + 1.0f, (float)(W - 1));
    float y1 = fminf(y0 + 1.0f, (float)(H - 1));
    float wx = xs - x0, wy = ys - y0;
    int x0i = (int)x0, x1i = (int)x1, y0i = (int)y0, y1i = (int)y1;
    float w00 = (1.0f - wx) * (1.0f - wy), w01 = wx * (1.0f - wy);
    float w10 = (1.0f - wx) * wy,          w11 = wx * wy;
    int i00 = y0i * W + x0i, i01 = y0i * W + x1i;
    int i10 = y1i * W + x0i, i11 = y1i * W + x1i;

    float ssd = 0.0f;
#pragma unroll 4
    for (int c = 0; c < C; ++c) {
      const float* qc = qb + (size_t)c * fs;
      const float* rc = rb + (size_t)c * fs;
      if (c + 4 < C) {  // uniform branch; prefetch 4 channels ahead
        __builtin_prefetch(qc + 4 * fs + i00, 0, 1);
        __builtin_prefetch(rc + 4 * fs + i00, 0, 1);
      }
      float qv = w00 * qc[i00] + w01 * qc[i01] + w10 * qc[i10] + w11 * qc[i11];
      float rv = w00 * rc[i00] + w01 * rc[i01] + w10 * rc[i10] + w11 * rc[i11];
      float d = qv - rv;
      ssd = fmaf(d, d, ssd);
    }
    float uv = w00 * ub[i00] + w01 * ub[i01] + w10 * ub[i10] + w11 * ub[i11];
    smem[m][l] = valid ? (ssd * uv * invN) : 0.0f;
  }
  __syncthreads();
  if (lane < 16) {
    float s = 0.0f;
    for (int j = 0; j < 16; ++j) s += smem[lane][j];
    atomicAdd(&res[(size_t)b * M_HYP + lane], s);
  }
}

// ---------------- kernel: LM noise step + pose compose ---------------------
__global__ void lm_update(float* __restrict__ Tcur, const float* __restrict__ res,
                          float* __restrict__ costs, float damping, int level, int it) {
  int idx = blockIdx.x * blockDim.x + threadIdx.x;
  if (idx >= B_SZ * M_HYP) return;
  int b = idx / M_HYP, m = idx % M_HYP;
  float r = res[idx];
  costs[idx] = r;  // final iteration leaves the reference's `costs`
  unsigned base = (((unsigned)(100 + level * 10 + it) * B_SZ + (unsigned)b) * M_HYP + (unsigned)m) * 6u;
  float step[6];
  for (int j = 0; j < 6; ++j)
    step[j] = -damping * r * hash_normal(base + (unsigned)j) * 0.01f;
  float delta[16];
  se3_exp(step, delta);
  float Told[16], Tnew[16];
  float* Tp = Tcur + (size_t)idx * 16;
  for (int i = 0; i < 16; ++i) Told[i] = Tp[i];
  mat4mul(Tnew, delta, Told);
  for (int i = 0; i < 16; ++i) Tp[i] = Tnew[i];
}

// ---------------- kernel: geodesic score + argmin + best pose --------------
__global__ void finalize(const float* __restrict__ Tcur, const float* __restrict__ Tpred,
                         const float* __restrict__ costs, float* __restrict__ outT) {
  int b = blockIdx.x * blockDim.x + threadIdx.x;
  if (b >= B_SZ) return;
  const float* Tp = Tpred + b * 16;
  float Ti[16];  // SE(3) inverse of T_pred
  Ti[0] = Tp[0]; Ti[1] = Tp[4]; Ti[2]  = Tp[8];
  Ti[4] = Tp[1]; Ti[5] = Tp[5]; Ti[6]  = Tp[9];
  Ti[8] = Tp[2]; Ti[9] = Tp[6]; Ti[10] = Tp[10];
  float tx = Tp[3], ty = Tp[7], tz = Tp[11];
  Ti[3]  = -(Ti[0] * tx + Ti[1] * ty + Ti[2]  * tz);
  Ti[7]  = -(Ti[4] * tx + Ti[5] * ty + Ti[6]  * tz);
  Ti[11] = -(Ti[8] * tx + Ti[9] * ty + Ti[10] * tz);
  Ti[12] = 0.0f; Ti[13] = 0.0f; Ti[14] = 0.0f; Ti[15] = 1.0f;

  float best = 1e30f;
  int bi = 0;
  for (int m = 0; m < M_HYP; ++m) {
    float Trel[16], xi[6];
    mat4mul(Trel, Tcur + ((size_t)b * M_HYP + m) * 16, Ti);
    se3_log(Trel, xi);
    float g = sqrtf(xi[0] * xi[0] + xi[1] * xi[1] + xi[2] * xi[2] +
                    xi[3] * xi[3] + xi[4] * xi[4] + xi[5] * xi[5]);
    float total = costs[b * M_HYP + m] + g;  // MOTION_LAMBDA = 1
    if (total < best) { best = total; bi = m; }
  }
  const float* Tb = Tcur + ((size_t)b * M_HYP + bi) * 16;
  for (int i = 0; i < 16; ++i) outT[b * 16 + i] = Tb[i];
}

// ---------------------------------------------------------------------------
extern "C" void kernel_launch(void* const* d_in, const int* in_sizes, int n_in,
                              void* d_out, int out_size, void* d_ws, size_t ws_size,
                              hipStream_t stream) {
  (void)in_sizes; (void)n_in; (void)out_size; (void)ws_size;
  const float* Tpred = (const float*)d_in[0];
  const float* geo   = (const float*)d_in[1];
  const float* Kin   = (const float*)d_in[2];
  const float* q[3]  = {(const float*)d_in[3], (const float*)d_in[4], (const float*)d_in[5]};
  const float* r[3]  = {(const float*)d_in[6], (const float*)d_in[7], (const float*)d_in[8]};
  const float* u[3]  = {(const float*)d_in[9], (const float*)d_in[10], (const float*)d_in[11]};

  float* ws   = (float*)d_ws;
  float* Tcur = ws;                                   // B*M*16
  float* res  = ws + (size_t)B_SZ * M_HYP * 16;       // B*M
  float* outT  = (float*)d_out;                       // B*16
  float* costs = outT + B_SZ * 16;                    // B*M

  gen_hyp<<<(B_SZ * M_HYP + 63) / 64, 64, 0, stream>>>(Tpred, Tcur);

  const int   Cs[3]    = {128, 64, 32};
  const int   Hs[3]    = {48, 96, 192};
  const int   Wl[3]    = {64, 128, 256};
  const int   iters[3] = {2, 3, 4};
  const float damp[3]  = {0.001f, 0.0005f, 0.00025f};  // LM_DAMPING * 0.5^level
  for (int level = 0; level < 3; ++level) {
    float scale = (float)(1 << level) * 0.25f;  // 1 / (4 / 2^level)
    for (int it = 0; it < iters[level]; ++it) {
      zero_res<<<(B_SZ * M_HYP + 255) / 256, 256, 0, stream>>>(res);
      dim3 grid((N_PTS + 15) / 16, B_SZ);
      residual_kernel<<<grid, 32, 0, stream>>>(Tcur, q[level], r[level], u[level],
                                               geo, Kin, res,
                                               Cs[level], Hs[level], Wl[level], scale);
      lm_update<<<(B_SZ * M_HYP + 63) / 64, 64, 0, stream>>>(Tcur, res, costs,
                                                             damp[level], level, it);
    }
  }
  finalize<<<1, 32, 0, stream>>>(Tcur, Tpred, costs, outT);
}